// NeighbourAggregation_50663434223640
// MI455X (gfx1250) — compile-verified
//
#include <hip/hip_runtime.h>
#include <stdint.h>

#define N_AGENTS 6144
#define TILE_J   512
#define BLK_I    16
#define NTHREADS 256
#define NWAVES   8

typedef float v2f __attribute__((ext_vector_type(2)));
typedef float v8f __attribute__((ext_vector_type(8)));

__global__ __launch_bounds__(NTHREADS)
void NeighbourAggregation_kernel(const float* __restrict__ states,
                                 const float* __restrict__ log_tau,
                                 float* __restrict__ out)
{
    __shared__ float  sStates[TILE_J * 8];        // 16 KB raw states tile (async staged)
    __shared__ float  sV[TILE_J][16];             // 32 KB feature rows [1, g, g^2, 0...]
    __shared__ float2 sPos[TILE_J];               //  4 KB pre-scaled positions
    __shared__ float  sRed[NWAVES][BLK_I][16];    //  8 KB cross-wave reduction

    const int tid  = threadIdx.x;
    const int lane = tid & 31;
    const int wave = tid >> 5;
    const int rowBase = blockIdx.x * BLK_I;

    // exp(-d/tau) = exp2(-(k*(a_i+b_i) - k*(a_j+b_j)) norm), with positions
    // pre-scaled by kh = 0.5 * log2(e) / tau so the inner loop is just
    // sub, sq-sum, sqrt, exp2(-x) (negate folds into v_exp src modifier).
    const float kh = 0.5f * 1.44269504088896f * __expf(-log_tau[0]);

    // Per-lane i-row setup (lanes l and l+16 both own row l&15, per WMMA A layout)
    const int   m   = lane & 15;
    const int   ig  = rowBase + m;
    const float4 srow = *(const float4*)(states + (size_t)ig * 8);
    const float pix = (srow.x + srow.z) * kh;     // pre-scaled position
    const float piy = (srow.y + srow.w) * kh;
    const int   k0  = (lane >> 4) * 2;            // K offset this lane feeds (0 or 2)

    v8f acc = {};

    for (int tile = 0; tile < N_AGENTS / TILE_J; ++tile) {
        const int tileBase = tile * TILE_J;
        __syncthreads();                          // previous tile fully consumed

        // ---- stage 2 raw state rows per thread: global -> LDS (async) ----
        {
            const int j0 = tid * 2;
            const float* g0 = states + (size_t)(tileBase + j0) * 8;
            const float* g1 = g0 + 8;
            uint32_t l0 = (uint32_t)(uintptr_t)&sStates[j0 * 8];
            uint32_t l1 = l0 + 32;
            asm volatile(
                "global_load_async_to_lds_b128 %0, %2, off\n\t"
                "global_load_async_to_lds_b128 %1, %3, off"
                :: "v"(l0), "v"(l0 + 16), "v"(g0), "v"(g0 + 4)
                : "memory");
            asm volatile(
                "global_load_async_to_lds_b128 %0, %2, off\n\t"
                "global_load_async_to_lds_b128 %1, %3, off"
                :: "v"(l1), "v"(l1 + 16), "v"(g1), "v"(g1 + 4)
                : "memory");
            asm volatile("s_wait_asynccnt 0" ::: "memory");

            // Build feature rows for the 2 rows this thread staged (reads own LDS rows)
            #pragma unroll
            for (int r = 0; r < 2; ++r) {
                const int j = j0 + r;
                const float4 sa = *(const float4*)&sStates[j * 8];
                const float4 sb = *(const float4*)&sStates[j * 8 + 4];
                float g0v = (sa.x + sa.z) * 0.5f;
                float g1v = (sa.y + sa.w) * 0.5f;
                float g2v = (sb.x + sb.z) * 0.5f;
                float g3v = (sb.y + sb.w) * 0.5f;
                float4* vr4 = (float4*)sV[j];
                vr4[0] = make_float4(1.0f, g0v, g1v, g2v);
                vr4[1] = make_float4(g3v, g0v * g0v, g1v * g1v, g2v * g2v);
                vr4[2] = make_float4(g3v * g3v, 0.f, 0.f, 0.f);
                vr4[3] = make_float4(0.f, 0.f, 0.f, 0.f);
                // pre-scaled positions (bitwise-identical op sequence to pix/piy)
                sPos[j] = make_float2((sa.x + sa.z) * kh, (sa.y + sa.w) * kh);
            }
        }
        __syncthreads();                          // tile features ready for all waves

        // ---- this wave covers 64 j's of the tile: 16 WMMA K=4 steps ----
        // NOTE: no diagonal masking here; w_ii = exp2(-sqrt(0)) = 1 exactly and
        // is subtracted once after the reduction.
        const int jw = wave * 64;
        #pragma unroll 8
        for (int step = 0; step < 16; ++step) {
            const int jl = jw + step * 4 + k0;    // tile-local j for this lane's K pair
            const float2 pj0 = sPos[jl];
            const float2 pj1 = sPos[jl + 1];
            float dx0 = pix - pj0.x, dy0 = piy - pj0.y;
            float dx1 = pix - pj1.x, dy1 = piy - pj1.y;
            float d0 = __builtin_amdgcn_sqrtf(dx0 * dx0 + dy0 * dy0);
            float d1 = __builtin_amdgcn_sqrtf(dx1 * dx1 + dy1 * dy1);
            float w0 = __builtin_amdgcn_exp2f(-d0);   // neg folds into v_exp
            float w1 = __builtin_amdgcn_exp2f(-d1);

            v2f a; a.x = w0;          a.y = w1;            // A[m][k0], A[m][k0+1]
            v2f b; b.x = sV[jl][m];   b.y = sV[jl + 1][m]; // B[k0][n], B[k0+1][n]

            acc = __builtin_amdgcn_wmma_f32_16x16x4_f32(
                      false, a, false, b, (short)0, acc, false, false);
        }
    }

    // ---- dump per-wave C tiles (16x16 f32 layout: VGPR r -> M=r / r+8) ----
    {
        const int mb = (lane < 16) ? 0 : 8;
        #pragma unroll
        for (int r = 0; r < 8; ++r)
            sRed[wave][mb + r][m] = acc[r];
    }
    __syncthreads();

    // ---- cross-wave sum + diagonal correction: one (m,n) element per thread ----
    {
        const int mm = tid >> 4, nn = tid & 15;
        float s = 0.0f;
        #pragma unroll
        for (int w = 0; w < NWAVES; ++w) s += sRed[w][mm][nn];

        // subtract the diagonal term: w_ii (=1) * feature_nn(row rowBase+mm)
        const int i = rowBase + mm;
        const float4 sa = *(const float4*)(states + (size_t)i * 8);
        const float4 sb = *(const float4*)(states + (size_t)i * 8 + 4);
        float g[4];
        g[0] = (sa.x + sa.z) * 0.5f;
        g[1] = (sa.y + sa.w) * 0.5f;
        g[2] = (sb.x + sb.z) * 0.5f;
        g[3] = (sb.y + sb.w) * 0.5f;
        float feat;
        if (nn == 0)       feat = 1.0f;
        else if (nn <= 4)  feat = g[nn - 1];
        else if (nn <= 8)  { float t = g[nn - 5]; feat = t * t; }
        else               feat = 0.0f;
        s -= feat;

        sRed[0][mm][nn] = s;                       // each thread owns its element
    }
    __syncthreads();

    // ---- finalize: 16 rows x 8 outputs ----
    if (tid < 128) {
        const int mm   = tid >> 3;
        const int comp = tid & 7;
        const int d    = comp & 3;
        const int i    = rowBase + mm;
        const float S  = sRed[0][mm][0];
        const float mv = sRed[0][mm][1 + d] / S;   // m_d = (alpha @ g)_d
        float o;
        if (comp < 4) {
            const int a = (d < 2) ? d : d + 2;
            const float gi = (states[(size_t)i*8 + a] + states[(size_t)i*8 + a + 2]) * 0.5f;
            o = gi - mv;                           // mu_d
        } else {
            const float q = sRed[0][mm][5 + d] / S;
            o = sqrtf(q - mv * mv + 1e-6f);        // sigma_d
        }
        out[(size_t)i * 8 + comp] = o;
    }
}

extern "C" void kernel_launch(void* const* d_in, const int* in_sizes, int n_in,
                              void* d_out, int out_size, void* d_ws, size_t ws_size,
                              hipStream_t stream) {
    (void)in_sizes; (void)n_in; (void)out_size; (void)d_ws; (void)ws_size;
    const float* states  = (const float*)d_in[0];
    const float* log_tau = (const float*)d_in[1];
    float* out = (float*)d_out;
    NeighbourAggregation_kernel<<<dim3(N_AGENTS / BLK_I), dim3(NTHREADS), 0, stream>>>(
        states, log_tau, out);
}